// GATLayer_61418032333370
// MI455X (gfx1250) — compile-verified
//
#include <hip/hip_runtime.h>

#define D 128
#define TILE_ROWS 16
#define LDS_PITCH 132  // 128 + 4 floats padding -> conflict-free column reads

typedef __attribute__((ext_vector_type(2))) float v2f;
typedef __attribute__((ext_vector_type(8))) float v8f;

__device__ __forceinline__ int f2oi(float f) {
  int i = __float_as_int(f);
  return (i >= 0) ? i : (i ^ 0x7FFFFFFF);
}
__device__ __forceinline__ float oi2f(int i) {
  return __int_as_float((i >= 0) ? i : (i ^ 0x7FFFFFFF));
}

// ---------------- init: agg = 0, denom = 0, m = -inf (ordered-int) ------------
__global__ void k_init(float* __restrict__ agg, float* __restrict__ denom,
                       int* __restrict__ m, int n) {
  int j = blockIdx.x * blockDim.x + threadIdx.x;
  if (j < n * D) agg[j] = 0.0f;
  if (j < n) {
    denom[j] = 0.0f;
    m[j] = f2oi(-__builtin_inff());
  }
}

// ---------------- QKV projection: q/k/v = feat @ W{q,k,v} + b ----------------
// Block: 256 threads (8 waves). Each block handles 16 node rows; each wave
// owns one 16-wide column tile and keeps three f32 accumulators (Q, K, V).
// Feature tile is staged global->LDS with the CDNA5 async-to-LDS path.
__global__ __launch_bounds__(256) void k_qkv(
    const float* __restrict__ feat,
    const float* __restrict__ Wq, const float* __restrict__ bq,
    const float* __restrict__ Wk, const float* __restrict__ bk,
    const float* __restrict__ Wv, const float* __restrict__ bv,
    float* __restrict__ q, float* __restrict__ k, float* __restrict__ v,
    int n) {
  __shared__ float lds[TILE_ROWS * LDS_PITCH];
  const int row0 = blockIdx.x * TILE_ROWS;
  const int t = threadIdx.x;

  // stage 16x128 fp32 tile: 512 float4 chunks, 2 per thread, direct to LDS.
  // chunk id c: row = c>>5 (32 float4/row), col4 = c&31.
  #pragma unroll
  for (int i = 0; i < 2; ++i) {
    int cid = t + i * 256;
    int r = cid >> 5, c4 = cid & 31;
    int row = row0 + r;
    if (row >= n) row = n - 1;  // clamp (tail rows never stored)
    unsigned ldsa = (unsigned)(uintptr_t)(&lds[r * LDS_PITCH + c4 * 4]);
    unsigned long long ga =
        (unsigned long long)(uintptr_t)(feat + (size_t)row * D + c4 * 4);
    asm volatile("global_load_async_to_lds_b128 %0, %1, off"
                 :
                 : "v"(ldsa), "v"(ga)
                 : "memory");
  }
  asm volatile("s_wait_asynccnt 0x0" ::: "memory");
  __syncthreads();

  const int lane = t & 31;
  const int wave = t >> 5;
  const int col0 = wave * 16;
  const int nn = lane & 15;   // N (and A-frag M) index within tile
  const int half = lane >> 4; // selects K pair {0,1} vs {2,3}

  v8f cq = {}; v8f ck = {}; v8f cv = {};
  #pragma unroll 4
  for (int kb = 0; kb < D; kb += 4) {
    const int ka = kb + half * 2;
    v2f a;
    a.x = lds[nn * LDS_PITCH + ka];
    a.y = lds[nn * LDS_PITCH + ka + 1];
    v2f b;
    b.x = Wq[(size_t)ka * D + col0 + nn];
    b.y = Wq[(size_t)(ka + 1) * D + col0 + nn];
    cq = __builtin_amdgcn_wmma_f32_16x16x4_f32(false, a, false, b, (short)0, cq, false, false);
    b.x = Wk[(size_t)ka * D + col0 + nn];
    b.y = Wk[(size_t)(ka + 1) * D + col0 + nn];
    ck = __builtin_amdgcn_wmma_f32_16x16x4_f32(false, a, false, b, (short)0, ck, false, false);
    b.x = Wv[(size_t)ka * D + col0 + nn];
    b.y = Wv[(size_t)(ka + 1) * D + col0 + nn];
    cv = __builtin_amdgcn_wmma_f32_16x16x4_f32(false, a, false, b, (short)0, cv, false, false);
  }

  const float biasq = bq[col0 + nn];
  const float biask = bk[col0 + nn];
  const float biasv = bv[col0 + nn];
  #pragma unroll
  for (int r = 0; r < 8; ++r) {
    int row = row0 + r + half * 8;  // D-matrix: VGPR r holds M=r (lanes 0-15), M=r+8 (16-31)
    if (row < n) {
      size_t off = (size_t)row * D + col0 + nn;
      q[off] = cq[r] + biasq;
      k[off] = ck[r] + biask;
      v[off] = cv[r] + biasv;
    }
  }
}

// ---------------- edge scores + segment max (one wave32 per edge) ------------
__global__ __launch_bounds__(256) void k_edge_score(
    const float* __restrict__ q, const float* __restrict__ k,
    const int* __restrict__ src, const int* __restrict__ dst,
    float* __restrict__ s, int* __restrict__ m, int e_cnt) {
  int gid = blockIdx.x * blockDim.x + threadIdx.x;
  int e = gid >> 5;
  if (e >= e_cnt) return;
  int lane = gid & 31;
  int d = dst[e], sr = src[e];
  float4 qa = ((const float4*)(q + (size_t)d * D))[lane];
  float4 ka = ((const float4*)(k + (size_t)sr * D))[lane];
  float p = qa.x * ka.x + qa.y * ka.y + qa.z * ka.z + qa.w * ka.w;
  #pragma unroll
  for (int off = 16; off > 0; off >>= 1) p += __shfl_xor(p, off, 32);
  if (lane == 0) {
    s[e] = p;
    atomicMax(m + d, f2oi(p));
  }
}

// ---------------- exp weights + weighted aggregation (wave per edge) ---------
__global__ __launch_bounds__(256) void k_edge_agg(
    const float* __restrict__ vv, const float* __restrict__ s,
    const int* __restrict__ src, const int* __restrict__ dst,
    const int* __restrict__ m, float* __restrict__ denom,
    float* __restrict__ agg, int e_cnt) {
  int gid = blockIdx.x * blockDim.x + threadIdx.x;
  int e = gid >> 5;
  if (e >= e_cnt) return;
  int lane = gid & 31;
  int d = dst[e], sr = src[e];
  float w = expf(s[e] - oi2f(m[d]));
  if (lane == 0) atomicAdd(denom + d, w);
  float4 vf = ((const float4*)(vv + (size_t)sr * D))[lane];
  float* ag = agg + (size_t)d * D + lane * 4;
  atomicAdd(ag + 0, w * vf.x);
  atomicAdd(ag + 1, w * vf.y);
  atomicAdd(ag + 2, w * vf.z);
  atomicAdd(ag + 3, w * vf.w);
}

// ---------------- out = relu((deg>0 ? agg/denom : v) @ Wl + bl) --------------
__global__ __launch_bounds__(256) void k_out(
    const float* __restrict__ agg, const float* __restrict__ denom,
    const float* __restrict__ v, const float* __restrict__ Wl,
    const float* __restrict__ bl, float* __restrict__ out, int n) {
  __shared__ float lds[TILE_ROWS * LDS_PITCH];
  const int row0 = blockIdx.x * TILE_ROWS;
  const int t = threadIdx.x;

  // staging computes (agg/denom | v) per element -> must go through VGPRs
  for (int i = 0; i < 8; ++i) {
    int lin = t + i * 256;
    int r = lin >> 7, c = lin & 127;
    int row = row0 + r;
    float val = 0.0f;
    if (row < n) {
      float dn = denom[row];
      size_t off = (size_t)row * D + c;
      val = (dn > 0.0f) ? (agg[off] / dn) : v[off];
    }
    lds[r * LDS_PITCH + c] = val;
  }
  __syncthreads();

  const int lane = t & 31;
  const int wave = t >> 5;
  const int col0 = wave * 16;
  const int nn = lane & 15;
  const int half = lane >> 4;

  v8f c = {};
  #pragma unroll 4
  for (int kb = 0; kb < D; kb += 4) {
    const int ka = kb + half * 2;
    v2f a;
    a.x = lds[nn * LDS_PITCH + ka];
    a.y = lds[nn * LDS_PITCH + ka + 1];
    v2f b;
    b.x = Wl[(size_t)ka * D + col0 + nn];
    b.y = Wl[(size_t)(ka + 1) * D + col0 + nn];
    c = __builtin_amdgcn_wmma_f32_16x16x4_f32(false, a, false, b, (short)0, c, false, false);
  }

  const float bias = bl[col0 + nn];
  #pragma unroll
  for (int r = 0; r < 8; ++r) {
    int row = row0 + r + half * 8;
    if (row < n) {
      out[(size_t)row * D + col0 + nn] = fmaxf(c[r] + bias, 0.0f);
    }
  }
}

extern "C" void kernel_launch(void* const* d_in, const int* in_sizes, int n_in,
                              void* d_out, int out_size, void* d_ws, size_t ws_size,
                              hipStream_t stream) {
  const float* feat = (const float*)d_in[0];
  const float* Wq = (const float*)d_in[1];
  const float* bq = (const float*)d_in[2];
  const float* Wk = (const float*)d_in[3];
  const float* bk = (const float*)d_in[4];
  const float* Wv = (const float*)d_in[5];
  const float* bv = (const float*)d_in[6];
  const float* Wl = (const float*)d_in[7];
  const float* bl = (const float*)d_in[8];
  const int* src = (const int*)d_in[9];
  const int* dst = (const int*)d_in[10];

  const int n = in_sizes[0] / D;   // 50000
  const int e_cnt = in_sizes[9];   // 800000
  float* out = (float*)d_out;

  // workspace carve-up (all fp32 words): q,k,v,agg [n*128 each], s [E],
  // denom [n], m [n ints]  -> ~106 MB total
  float* q = (float*)d_ws;
  float* k = q + (size_t)n * D;
  float* v = k + (size_t)n * D;
  float* agg = v + (size_t)n * D;
  float* s = agg + (size_t)n * D;
  float* denom = s + e_cnt;
  int* m = (int*)(denom + n);

  const int nblk = (n + TILE_ROWS - 1) / TILE_ROWS;
  const int egrid = (int)(((long long)e_cnt * 32 + 255) / 256);

  k_init<<<(n * D + 255) / 256, 256, 0, stream>>>(agg, denom, m, n);
  k_qkv<<<nblk, 256, 0, stream>>>(feat, Wq, bq, Wk, bk, Wv, bv, q, k, v, n);
  k_edge_score<<<egrid, 256, 0, stream>>>(q, k, src, dst, s, m, e_cnt);
  k_edge_agg<<<egrid, 256, 0, stream>>>(v, s, src, dst, m, denom, agg, e_cnt);
  k_out<<<nblk, 256, 0, stream>>>(agg, denom, v, Wl, bl, out, n);
}